// NetM_84456236908650
// MI455X (gfx1250) — compile-verified
//
#include <hip/hip_runtime.h>

#define HH  480
#define WWp 640
#define HPc 30
#define WPc 40
#define NBc 128
#define PPc (HPc * WPc)   // 1200 points per image

// ---------------------------------------------------------------------------
// CDNA5 async LDS->global store path (gfx1250 only), guarded so both the host
// pass and any toolchain lacking the builtin fall back to plain B128 stores.
// Probe result (round 1): builtin's param 0 is AS(1) pointer to int32x4.
// ---------------------------------------------------------------------------
#if defined(__AMDGCN__) && defined(__gfx1250__) && \
    __has_builtin(__builtin_amdgcn_global_store_async_from_lds_b128)
#define USE_ASYNC_ZERO 1
#else
#define USE_ASYNC_ZERO 0
#endif

#if USE_ASYNC_ZERO
typedef int v4i __attribute__((vector_size(4 * sizeof(int))));
typedef __attribute__((address_space(1))) v4i* gv4i_p;
typedef __attribute__((address_space(3))) v4i* lv4i_p;
typedef __attribute__((address_space(1))) void* gvoid_p;
typedef __attribute__((address_space(3))) void* lvoid_p;
#endif

// Zero-fill the dense sparse_depth region (n4 float4's = 157 MB). Pure
// bandwidth: ~7us floor at 23.3 TB/s. Async engine streams zeros from a 4KB
// LDS tile (ASYNCcnt-tracked), freeing the VALU/VGPR writeback path.
__global__ void __launch_bounds__(256)
zero_fill_kernel(float4* __restrict__ out, long n4) {
  const long tid    = (long)blockIdx.x * blockDim.x + threadIdx.x;
  const long stride = (long)gridDim.x * blockDim.x;
#if USE_ASYNC_ZERO
  __shared__ __align__(16) float4 lz[256];
  lz[threadIdx.x] = make_float4(0.0f, 0.0f, 0.0f, 0.0f);
  __syncthreads();  // DS writes visible before the async engine reads LDS
  lv4i_p lsrc = (lv4i_p)(lvoid_p)&lz[threadIdx.x];
  for (long i = tid; i < n4; i += stride) {
    gv4i_p gdst = (gv4i_p)(gvoid_p)(out + i);
    __builtin_amdgcn_global_store_async_from_lds_b128(gdst, lsrc,
                                                      /*offset=*/0, /*cpol=*/0);
  }
#if __has_builtin(__builtin_amdgcn_s_wait_asynccnt)
  __builtin_amdgcn_s_wait_asynccnt(0);  // drain ASYNCcnt before s_endpgm
#endif
#else
  const float4 z = make_float4(0.0f, 0.0f, 0.0f, 0.0f);
  for (long i = tid; i < n4; i += stride) out[i] = z;
#endif
}

// One thread per pooled point: grid output, bilinear gather (L2-resident
// images), scatter-add via no-return global_atomic_add_f32.
__global__ void __launch_bounds__(256)
sample_scatter_kernel(const float* __restrict__ xy,
                      const float* __restrict__ depth,
                      float* __restrict__ out) {
  const int idx = blockIdx.x * blockDim.x + threadIdx.x;
  if (idx >= NBc * PPc) return;
  const int b = idx / PPc;
  const int p = idx - b * PPc;

  // pooled_xy_tensor layout: [B, 2, HP, WP]; channel 0 = x, channel 1 = y
  const float x = xy[(size_t)b * 2 * PPc + p];
  const float y = xy[(size_t)b * 2 * PPc + PPc + p];

  // grid_to_sample = ((2*x - W)/W, (2*y - H)/H)
  const float gw = fmaf(x, 2.0f / WWp, -1.0f);
  const float gh = fmaf(y, 2.0f / HH, -1.0f);

  // second output: [B, HP, WP, 2] appended after sparse_depth
  ((float2*)(out + (size_t)NBc * HH * WWp))[idx] = make_float2(gw, gh);

  // grid_sample, align_corners=True: map back to pixel coords
  const float xs = (gw + 1.0f) * 0.5f * (float)(WWp - 1);
  const float ys = (gh + 1.0f) * 0.5f * (float)(HH - 1);
  const float x0f = floorf(xs), y0f = floorf(ys);
  const float wx1 = xs - x0f, wy1 = ys - y0f;
  const float wx0 = 1.0f - wx1, wy0 = 1.0f - wy1;
  const int x0 = (int)x0f, y0 = (int)y0f;
  const int x1 = x0 + 1, y1 = y0 + 1;

  const bool vx0 = (x0 >= 0) && (x0 <= WWp - 1);
  const bool vx1 = (x1 >= 0) && (x1 <= WWp - 1);
  const bool vy0 = (y0 >= 0) && (y0 <= HH - 1);
  const bool vy1 = (y1 >= 0) && (y1 <= HH - 1);
  const int xc0 = min(max(x0, 0), WWp - 1), xc1 = min(max(x1, 0), WWp - 1);
  const int yc0 = min(max(y0, 0), HH - 1),  yc1 = min(max(y1, 0), HH - 1);

  const float* __restrict__ img = depth + (size_t)b * HH * WWp;
  const float v00 = (vy0 && vx0) ? img[yc0 * WWp + xc0] : 0.0f;
  const float v01 = (vy0 && vx1) ? img[yc0 * WWp + xc1] : 0.0f;
  const float v10 = (vy1 && vx0) ? img[yc1 * WWp + xc0] : 0.0f;
  const float v11 = (vy1 && vx1) ? img[yc1 * WWp + xc1] : 0.0f;

  const float val = v00 * wy0 * wx0 + v01 * wy0 * wx1 +
                    v10 * wy1 * wx0 + v11 * wy1 * wx1;

  // rows/cols: round-half-even (rintf matches jnp.round), then clip
  const int r = (int)fminf(fmaxf(rintf(y), 0.0f), (float)(HH - 1));
  const int c = (int)fminf(fmaxf(rintf(x), 0.0f), (float)(WWp - 1));

  atomicAdd(out + (size_t)b * HH * WWp + (size_t)r * WWp + c, val);
}

extern "C" void kernel_launch(void* const* d_in, const int* in_sizes, int n_in,
                              void* d_out, int out_size, void* d_ws, size_t ws_size,
                              hipStream_t stream) {
  (void)in_sizes; (void)n_in; (void)out_size; (void)d_ws; (void)ws_size;
  const float* xy    = (const float*)d_in[0];  // pooled_xy_tensor [128,2,30,40]
  const float* depth = (const float*)d_in[1];  // image_depth     [128,1,480,640]
  float* out = (float*)d_out;                  // [sparse_depth | grid_to_sample]

  const long n4 = (long)NBc * HH * WWp / 4;    // 9,830,400 float4 stores
  // 4800 blocks x 256 thr -> exactly 8 B128 stores per thread
  zero_fill_kernel<<<4800, 256, 0, stream>>>((float4*)out, n4);

  const int npts = NBc * PPc;                  // 153,600
  sample_scatter_kernel<<<(npts + 255) / 256, 256, 0, stream>>>(xy, depth, out);
}